// PlainGCN_83004537963208
// MI455X (gfx1250) — compile-verified
//
#include <hip/hip_runtime.h>

// Problem constants (match reference)
#define NPTS   12000
#define CF     64
#define KNN    16
#define TILE   16
#define NTILES (NPTS / TILE)   // 750
#define WPB    4               // waves per block in knn kernel
#define BROW   68              // padded LDS row stride (floats) for B tiles
#define BIGD   1.0e9f

typedef __attribute__((ext_vector_type(2))) float v2f;
typedef __attribute__((ext_vector_type(8))) float v8f;
typedef __attribute__((ext_vector_type(4))) unsigned int u32x4;
typedef __attribute__((ext_vector_type(8))) int i32x8;
typedef __attribute__((ext_vector_type(4))) int i32x4;

__device__ __forceinline__ v8f wmma_f32_k4(v2f a, v2f b, v8f c) {
  return __builtin_amdgcn_wmma_f32_16x16x4_f32(false, a, false, b, (short)0, c,
                                               false, false);
}

// TDM: DMA a 16x64 f32 tile (rows of `feat`) into LDS at byte offset ldsOff,
// inserting 4 pad DWORDs after every 64 DWORDs (row stride 68 floats in LDS).
__device__ __forceinline__ void tdm_load_tile(const float* gsrc,
                                              unsigned int ldsOff) {
  unsigned long long ga = (unsigned long long)(uintptr_t)gsrc;
  u32x4 g0;
  g0.x = 1u;                                        // count=1, user mode
  g0.y = ldsOff;                                    // lds_addr (bytes)
  g0.z = (unsigned int)ga;                          // global_addr[31:0]
  g0.w = (unsigned int)((ga >> 32) & 0x1FFFFFFu)    // global_addr[56:32]
         | (2u << 30);                              // type = 2 ("image")
  i32x8 g1;
  g1[0] = (int)((2u << 16)     // data_size = 4 bytes
                | (1u << 20)   // pad_enable
                | (5u << 22)   // pad_interval: every 64 DWORDs
                | (3u << 25)); // pad_amount: 4 DWORDs
  g1[1] = (int)(64u << 16);    // tensor_dim0 = 64 (low 16 bits)
  g1[2] = (int)(((unsigned int)NPTS & 0xFFFFu) << 16);  // tensor_dim1 lo16
  g1[3] = (int)(64u << 16);    // tile_dim0 = 64
  g1[4] = 16;                  // tile_dim1 = 16, tile_dim2 = 0
  g1[5] = 64;                  // tensor_dim0_stride = 64
  g1[6] = 0;
  g1[7] = 0;
  i32x4 gz4 = {0, 0, 0, 0};
  i32x8 gz8 = {0, 0, 0, 0, 0, 0, 0, 0};
  // 6-arg toolchain variant: (g0, g1, g2, g3, extra, cpol)
  __builtin_amdgcn_tensor_load_to_lds(g0, g1, gz4, gz4, gz8, 0);
}

__device__ __forceinline__ void mem_fence_compiler() {
  asm volatile("" ::: "memory");
}

// ---------------------------------------------------------------------------
// Prep: batch ids (int) + copy pillar features into working buffer
// ---------------------------------------------------------------------------
__global__ void prep_kernel(const float* __restrict__ vox,
                            const float* __restrict__ pillar,
                            float* __restrict__ feat, int* __restrict__ bat) {
  int t = blockIdx.x * blockDim.x + threadIdx.x;
  if (t < NPTS) bat[t] = (int)vox[t * 4];
  if (t < NPTS * CF) feat[t] = pillar[t];
}

// ---------------------------------------------------------------------------
// Squared row norms
// ---------------------------------------------------------------------------
__global__ void norm_kernel(const float* __restrict__ feat,
                            float* __restrict__ sq) {
  int i = blockIdx.x * blockDim.x + threadIdx.x;
  if (i >= NPTS) return;
  const float4* p = (const float4*)(feat + i * CF);
  float s = 0.0f;
#pragma unroll
  for (int c = 0; c < CF / 4; ++c) {
    float4 v = p[c];
    s += v.x * v.x + v.y * v.y + v.z * v.z + v.w * v.w;
  }
  sq[i] = s;
}

// ---------------------------------------------------------------------------
// Fused distance GEMM (WMMA f32 16x16x4) + running top-K selection.
// 4 waves/block, one wave per 16-row tile. B tiles staged in LDS by the
// Tensor Data Mover (double-buffered, s_wait_tensorcnt), WMMAs back-to-back.
// ---------------------------------------------------------------------------
__global__ __launch_bounds__(WPB * 32) void knn_kernel(
    const float* __restrict__ feat, const float* __restrict__ sq,
    const int* __restrict__ bat, int* __restrict__ idxOut) {
  const int lane = threadIdx.x & 31;
  const int wid  = threadIdx.x >> 5;
  const int m    = lane & 15;
  const int hi   = lane >> 4;
  const int koff = hi * 2;
  const int rowTile = blockIdx.x * WPB + wid;

  __shared__ float btile[WPB][2][TILE * BROW];  // TDM-staged B tiles (padded)
  __shared__ float ldsD[WPB][TILE * TILE];      // distance tile, row-major
  __shared__ float sD[WPB][32 * KNN];           // per-lane sorted lists
  __shared__ int   sI[WPB][32 * KNN];

  if (rowTile >= NTILES) return;  // no block-wide barriers used below
  const int rowBase = rowTile * TILE;

  const unsigned int bOff0 =
      (unsigned int)(uintptr_t)(void*)&btile[wid][0][0];
  const unsigned int bOff1 =
      (unsigned int)(uintptr_t)(void*)&btile[wid][1][0];

  // Preload A fragments: afr[kk] = feat[row m][kk*4+koff .. +1]
  v2f afr[16];
  {
    const float* ap = feat + (rowBase + m) * CF + koff;
#pragma unroll
    for (int kk = 0; kk < 16; ++kk) {
      v2f a;
      a.x = ap[kk * 4 + 0];
      a.y = ap[kk * 4 + 1];
      afr[kk] = a;
    }
  }

  // Row-side norms / batch ids for the 8 accumulator rows this lane owns.
  float sqr[8];
  int   batr[8];
#pragma unroll
  for (int v = 0; v < 8; ++v) {
    int r = rowBase + hi * 8 + v;
    sqr[v]  = sq[r];
    batr[v] = bat[r];
  }

  // Per-lane sorted top-K (lane L and L+16 jointly cover row L&15).
  float bd[KNN];
  int   bi[KNN];
#pragma unroll
  for (int k = 0; k < KNN; ++k) { bd[k] = BIGD; bi[k] = 0; }

  // Prologue: DMA tile 0 into buffer 0.
  tdm_load_tile(feat, bOff0);

  for (int ct = 0; ct < NTILES; ++ct) {
    // Issue DMA for next tile (wraps on last iteration; harmless).
    const int nb = (ct + 1 < NTILES) ? (ct + 1) : 0;
    tdm_load_tile(feat + (size_t)nb * TILE * CF,
                  ((ct + 1) & 1) ? bOff1 : bOff0);

    // Wait until the current tile's DMA is complete (<=1 outstanding).
    __builtin_amdgcn_s_wait_tensorcnt(1);
    mem_fence_compiler();

    const int cBase = ct * TILE;
    const int cCol  = cBase + m;
    const float* bt = &btile[wid][ct & 1][0];

    // Bulk-read B fragments from LDS, then chain 16 WMMAs back-to-back.
    v2f bfr[16];
#pragma unroll
    for (int kk = 0; kk < 16; ++kk) {
      const float* bp = bt + m * BROW + kk * 4 + koff;
      v2f b;
      b.x = bp[0];
      b.y = bp[1];
      bfr[kk] = b;
    }
    v8f acc = {};
#pragma unroll
    for (int kk = 0; kk < 16; ++kk) acc = wmma_f32_k4(afr[kk], bfr[kk], acc);

    const float sqc  = sq[cCol];
    const int   batc = bat[cCol];

    // Distances + masking, transpose through LDS (wave-local, LDS in-order).
#pragma unroll
    for (int v = 0; v < 8; ++v) {
      int r    = hi * 8 + v;
      int gRow = rowBase + r;
      float d  = sqr[v] + sqc - 2.0f * acc[v];
      if (batr[v] != batc || gRow == cCol) d = BIGD;
      ldsD[wid][r * TILE + m] = d;
    }
    mem_fence_compiler();

    // Selection: lane handles row m, columns [hi*8, hi*8+8).
#pragma unroll
    for (int c = 0; c < 8; ++c) {
      int col  = hi * 8 + c;
      float d  = ldsD[wid][m * TILE + col];
      int   gi = cBase + col;
      if (d < bd[KNN - 1]) {
#pragma unroll
        for (int k = 0; k < KNN; ++k) {
          if (d < bd[k]) {
            float td = bd[k]; int ti = bi[k];
            bd[k] = d; bi[k] = gi;
            d = td; gi = ti;
          }
        }
      }
    }
    mem_fence_compiler();
  }

  // Merge the two half-lists per row (lane L with lane L+16).
#pragma unroll
  for (int k = 0; k < KNN; ++k) {
    sD[wid][lane * KNN + k] = bd[k];
    sI[wid][lane * KNN + k] = bi[k];
  }
  mem_fence_compiler();
  if (lane < 16) {
    int p0 = 0, p1 = 0;
    int* orow = idxOut + (rowBase + lane) * KNN;
#pragma unroll
    for (int k = 0; k < KNN; ++k) {
      float d0 = sD[wid][lane * KNN + p0];
      float d1 = sD[wid][(lane + 16) * KNN + p1];
      bool take0 = d0 <= d1;
      orow[k] = take0 ? sI[wid][lane * KNN + p0] : sI[wid][(lane + 16) * KNN + p1];
      p0 += take0 ? 1 : 0;
      p1 += take0 ? 0 : 1;
    }
  }
}

// ---------------------------------------------------------------------------
// Xt = feat @ W[0:64,:] + b ; Xb = feat @ W[64:128,:]  (WMMA f32 16x16x4)
// grid = (750, 8): blockIdx.y -> {half, colTile}
// ---------------------------------------------------------------------------
__global__ __launch_bounds__(32) void lin_kernel(
    const float* __restrict__ feat, const float* __restrict__ W /*128x64*/,
    const float* __restrict__ bias, float* __restrict__ Xt,
    float* __restrict__ Xb) {
  const int lane = threadIdx.x;
  const int m    = lane & 15;
  const int hi   = lane >> 4;
  const int koff = hi * 2;
  const int rowBase = blockIdx.x * TILE;
  const int half  = blockIdx.y >> 2;     // 0 -> Xt, 1 -> Xb
  const int ctile = blockIdx.y & 3;      // output col tile
  const float* Wp = W + half * CF * CF;

  const int n = ctile * TILE + m;

  v2f afr[16];
  v2f bfr[16];
  {
    const float* ap = feat + (rowBase + m) * CF + koff;
#pragma unroll
    for (int kk = 0; kk < 16; ++kk) {
      v2f a;
      a.x = ap[kk * 4 + 0];
      a.y = ap[kk * 4 + 1];
      afr[kk] = a;
      int k0 = kk * 4 + koff;
      v2f b;
      b.x = Wp[(k0 + 0) * CF + n];
      b.y = Wp[(k0 + 1) * CF + n];
      bfr[kk] = b;
    }
  }

  v8f acc = {};
#pragma unroll
  for (int kk = 0; kk < 16; ++kk) acc = wmma_f32_k4(afr[kk], bfr[kk], acc);

  float* out = half ? Xb : Xt;
  const float bn = half ? 0.0f : bias[n];
#pragma unroll
  for (int v = 0; v < 8; ++v) {
    int r = rowBase + hi * 8 + v;
    out[r * CF + n] = acc[v] + bn;
  }
}

// ---------------------------------------------------------------------------
// feat_new[i,c] = max_j relu(Xt[i,c] - Xb[i,c] + Xb[idx[i,j],c])
// ---------------------------------------------------------------------------
__global__ void agg_kernel(const float* __restrict__ Xt,
                           const float* __restrict__ Xb,
                           const int* __restrict__ idx,
                           float* __restrict__ outFeat) {
  int t = blockIdx.x * blockDim.x + threadIdx.x;
  if (t >= NPTS * CF) return;
  int i = t >> 6;
  int c = t & (CF - 1);
  float base = Xt[t] - Xb[t];
  const int* row = idx + i * KNN;
  float mx = 0.0f;  // relu() >= 0, so 0 is the identity for max-of-relu
#pragma unroll
  for (int j = 0; j < KNN; ++j) {
    float h = base + Xb[row[j] * CF + c];
    mx = fmaxf(mx, h);
  }
  outFeat[t] = mx;
}

// ---------------------------------------------------------------------------
// out = pillar + feat
// ---------------------------------------------------------------------------
__global__ void add_kernel(const float* __restrict__ pillar,
                           const float* __restrict__ feat,
                           float* __restrict__ out) {
  int t = blockIdx.x * blockDim.x + threadIdx.x;
  if (t < NPTS * CF) out[t] = pillar[t] + feat[t];
}

// ---------------------------------------------------------------------------
extern "C" void kernel_launch(void* const* d_in, const int* in_sizes, int n_in,
                              void* d_out, int out_size, void* d_ws,
                              size_t ws_size, hipStream_t stream) {
  const float* vox    = (const float*)d_in[0];  // [N,4]
  const float* pillar = (const float*)d_in[1];  // [N,64]
  const float* Ws[3]  = {(const float*)d_in[2], (const float*)d_in[4],
                         (const float*)d_in[6]};
  const float* bs[3]  = {(const float*)d_in[3], (const float*)d_in[5],
                         (const float*)d_in[7]};

  char* ws = (char*)d_ws;
  float* featA = (float*)ws;                 ws += (size_t)NPTS * CF * 4;
  float* featB = (float*)ws;                 ws += (size_t)NPTS * CF * 4;
  float* Xt    = (float*)ws;                 ws += (size_t)NPTS * CF * 4;
  float* Xb    = (float*)ws;                 ws += (size_t)NPTS * CF * 4;
  float* sq    = (float*)ws;                 ws += (size_t)NPTS * 4;
  int*   bat   = (int*)ws;                   ws += (size_t)NPTS * 4;
  int*   idx   = (int*)ws;                   ws += (size_t)NPTS * KNN * 4;

  const int NC = NPTS * CF;
  prep_kernel<<<(NC + 255) / 256, 256, 0, stream>>>(vox, pillar, featA, bat);

  float* cur = featA;
  float* nxt = featB;
  for (int l = 0; l < 3; ++l) {
    norm_kernel<<<(NPTS + 255) / 256, 256, 0, stream>>>(cur, sq);
    knn_kernel<<<(NTILES + WPB - 1) / WPB, WPB * 32, 0, stream>>>(cur, sq, bat,
                                                                  idx);
    dim3 g(NTILES, 8);
    lin_kernel<<<g, 32, 0, stream>>>(cur, Ws[l], bs[l], Xt, Xb);
    agg_kernel<<<(NC + 255) / 256, 256, 0, stream>>>(Xt, Xb, idx, nxt);
    float* t = cur; cur = nxt; nxt = t;
  }
  add_kernel<<<(NC + 255) / 256, 256, 0, stream>>>(pillar, cur, (float*)d_out);
}